// PercentageEliminationLoss_22960895164702
// MI455X (gfx1250) — compile-verified
//
#include <hip/hip_runtime.h>
#include <hip/hip_bf16.h>

typedef __attribute__((ext_vector_type(2))) float v2f;
typedef __attribute__((ext_vector_type(8))) float v8f;

#define B_      128
#define N_      4096
#define K_      256
#define NCHUNK  256
#define NBLK    (N_ / NCHUNK)   // 16 n-chunks per batch
#define MARGIN_F 0.01f

struct PelAccum { float loss; unsigned int pairs; };

__global__ void pel_init(PelAccum* acc) {
    acc->loss  = 0.0f;
    acc->pairs = 0u;
}

__global__ __launch_bounds__(256)
void pel_pair(const float* __restrict__ scores,
              const int* __restrict__ idx,
              const unsigned char* __restrict__ mask,
              PelAccum* __restrict__ acc) {
    __shared__ float         se[K_];       // eliminated scores (-1e30 if invalid)
    __shared__ unsigned char flag[NCHUNK]; // eliminated-set flags for this n-window
    __shared__ float         part[256];    // per-thread partial sums
    __shared__ unsigned int  s_nvalid, s_nsurv;

    const int t  = threadIdx.x;
    const int b  = blockIdx.x / NBLK;
    const int n0 = (blockIdx.x % NBLK) * NCHUNK;

    if (t == 0) { s_nvalid = 0u; s_nsurv = 0u; }
    flag[t] = 0;
    __syncthreads();

    // Gather eliminated entry t; mark elimination set within this window.
    int ki = idx[b * K_ + t] & (N_ - 1);
    bool ev = mask[b * N_ + ki] != 0;
    se[t] = ev ? scores[b * N_ + ki] : -1.0e30f;   // invalid -> relu term always 0
    if (ev) atomicAdd(&s_nvalid, 1u);
    int rel = ki - n0;
    if ((unsigned)rel < (unsigned)NCHUNK) flag[rel] = 1;  // benign races
    __syncthreads();

    // Survivor test for n = n0 + t.
    const int n = n0 + t;
    bool msk  = mask[b * N_ + n] != 0;
    bool surv = msk && (flag[t] == 0);
    float sn  = scores[b * N_ + n];
    sn = surv ? sn : 1.0e30f;                      // non-survivor -> all terms 0
    if (surv) atomicAdd(&s_nsurv, 1u);

    // Branch-free hinge accumulation over all K eliminated entries.
    float a = 0.0f;
    const float base = MARGIN_F - sn;
    #pragma unroll 8
    for (int k = 0; k < K_; ++k) {
        a += fmaxf(se[k] + base, 0.0f);
    }
    part[t] = a;
    __syncthreads();

    // Block reduction of 256 partials by wave 0 using f32 WMMA with an
    // all-ones A matrix: D = ones(16x4) x B(4x16) + C accumulates column
    // sums; chaining 4 WMMAs covers all 256 values. With A == 1, every
    // row of D equals the column-sum vector, so summing D's first VGPR
    // across all 32 lanes yields exactly 2x the grand total.
    if (t < 32) {   // uniform per wave -> EXEC all-ones inside (WMMA requirement)
        v2f ones; ones[0] = 1.0f; ones[1] = 1.0f;
        v8f c = {};
        #pragma unroll
        for (int chunk = 0; chunk < 4; ++chunk) {
            v2f bm;
            bm[0] = part[chunk * 64 + t];
            bm[1] = part[chunk * 64 + 32 + t];
            // (neg_a, A, neg_b, B, c_mod, C, reuse_a, reuse_b)
            c = __builtin_amdgcn_wmma_f32_16x16x4_f32(
                    false, ones, false, bm, (short)0, c, false, false);
        }
        float v = c[0];
        #pragma unroll
        for (int s = 16; s > 0; s >>= 1) v += __shfl_xor(v, s, 32);
        if (t == 0) {
            atomicAdd(&acc->loss, 0.5f * v);
            atomicAdd(&acc->pairs, s_nvalid * s_nsurv);  // separable pair count
        }
    }
}

__global__ void pel_final(const PelAccum* __restrict__ acc,
                          float* __restrict__ out) {
    unsigned int p = acc->pairs;
    out[0] = acc->loss / (float)(p == 0u ? 1u : p);
}

extern "C" void kernel_launch(void* const* d_in, const int* in_sizes, int n_in,
                              void* d_out, int out_size, void* d_ws, size_t ws_size,
                              hipStream_t stream) {
    const float*         scores = (const float*)d_in[0];
    const int*           idx    = (const int*)d_in[1];          // int per harness dtype rule
    const unsigned char* mask   = (const unsigned char*)d_in[2]; // bool -> u8
    PelAccum* acc = (PelAccum*)d_ws;
    float*    out = (float*)d_out;

    hipLaunchKernelGGL(pel_init,  dim3(1),          dim3(1),   0, stream, acc);
    hipLaunchKernelGGL(pel_pair,  dim3(B_ * NBLK),  dim3(256), 0, stream,
                       scores, idx, mask, acc);
    hipLaunchKernelGGL(pel_final, dim3(1),          dim3(1),   0, stream, acc, out);
}